// CIN_38311108281202
// MI455X (gfx1250) — compile-verified
//
#include <hip/hip_runtime.h>

// Problem constants (from reference)
#define BATCH 2048
#define MF    32      // num fields (x0 rows)
#define DF    32      // embedding dim

typedef __attribute__((ext_vector_type(2))) float v2f;
typedef __attribute__((ext_vector_type(8))) float v8f;
typedef __attribute__((ext_vector_type(4))) int   v4i;

#ifndef __has_builtin
#define __has_builtin(x) 0
#endif

#if __has_builtin(__builtin_amdgcn_global_load_async_to_lds_b128) && \
    __has_builtin(__builtin_amdgcn_s_wait_asynccnt)
#define HAVE_ASYNC 1
#else
#define HAVE_ASYNC 0
#endif

#define AS1 __attribute__((address_space(1)))
#define AS3 __attribute__((address_space(3)))

__device__ __forceinline__ void async_wait_all()
{
#if HAVE_ASYNC
    __builtin_amdgcn_s_wait_asynccnt(0);
#endif
}

// One CIN layer:  xout[b,o,d] = sum_{m,h} x0[b,m,d] * xin[b,h,d] * Wf[m*H+h, o]
// Per batch this is Out[OxD] = W^T[OxK] * Z[KxD], K = MF*H,
// Z[p,d] = x0[b, p>>LOG2H, d] * xin[b, p&(H-1), d] built on the fly from LDS.
//
// Wave = (batch-in-block, o-pair, d-tile): two accumulators (o0, o0+16) share
// one Z fragment -> Z build cost amortized over 2 WMMAs. W chunks are
// double-buffered in LDS via CDNA5 async-to-LDS loads (ASYNCcnt).
// All configs: 8 waves = 256 threads.
template<int H, int O, int LOG2H, int BT, int KC, bool XEQ>
__global__ __launch_bounds__(256)
void cin_layer_kernel(const float* __restrict__ x0,
                      const float* __restrict__ xin,
                      const float* __restrict__ Wf,
                      float* __restrict__ xout,     // [B,O,DF] or nullptr
                      float* __restrict__ finals,   // [B,224], atomic accumulate
                      int foff)
{
    constexpr int K    = MF * H;
    constexpr int NCH  = K / KC;              // number of W chunks
    constexpr int WPB  = (O / 32) * 2;        // waves per batch (o-pairs x d-tiles)
    constexpr int NW   = BT * WPB;            // waves per block
    constexpr int NTHR = NW * 32;
    constexpr int E4   = (KC * O) / 4;        // float4 elements per W chunk
    constexpr int E4PT = E4 / NTHR;           // per-thread copies (exact)
    static_assert(NTHR == 256, "block is 8 waves");
    static_assert(E4 % NTHR == 0, "copy divides");
    static_assert(K % KC == 0, "chunks divide");

    __shared__ alignas(16) float lds_x0[BT * MF * DF];
    __shared__ alignas(16) float lds_x [XEQ ? 4 : BT * H * DF];
    __shared__ alignas(16) float lds_w [2][KC * O];

    const int tid = threadIdx.x;
    const int b0  = blockIdx.x * BT;

    // Stage activations for the BT batches (coalesced).
    for (int i = tid; i < BT * MF * DF; i += NTHR)
        lds_x0[i] = x0[(size_t)b0 * MF * DF + i];
    if (!XEQ)
        for (int i = tid; i < BT * H * DF; i += NTHR)
            lds_x[i] = xin[(size_t)b0 * H * DF + i];

    // Cooperative W-chunk stage (async direct-to-LDS when available).
    auto stage = [&](int ci) {
        const float* gsrc = Wf + (size_t)ci * KC * O;
        float*       ldst = lds_w[ci & 1];
#pragma unroll
        for (int j = 0; j < E4PT; ++j) {
            const int idx = (tid + j * NTHR) * 4;
#if HAVE_ASYNC
            __builtin_amdgcn_global_load_async_to_lds_b128(
                (AS1 v4i*)(gsrc + idx), (AS3 v4i*)(ldst + idx), 0, 0);
#else
            *(float4*)(ldst + idx) = *(const float4*)(gsrc + idx);
#endif
        }
    };

    const int wave = tid >> 5;
    const int lane = tid & 31;
    const int bb   = wave / WPB;              // batch within block
    const int wl   = wave % WPB;
    const int o0   = (wl >> 1) * 32;          // o-pair base: tiles o0 and o0+16
    const int d0   = (wl & 1) * 16;           // embedding-col tile
    const int half = lane >> 4;
    const int lm   = lane & 15;

    const float* xb0 = &lds_x0[bb * MF * DF];
    const float* xbx = XEQ ? xb0 : &lds_x[bb * H * DF];

    v8f c0 = {0.f,0.f,0.f,0.f,0.f,0.f,0.f,0.f};
    v8f c1 = c0;

    stage(0);   // prologue: chunk 0 in flight

    for (int ci = 0; ci < NCH; ++ci) {
        async_wait_all();        // my chunk-ci loads complete (in-order counter)
        __syncthreads();         // (A) everyone's chunk ci + activations ready
        if (ci + 1 < NCH) stage(ci + 1);   // overlap next chunk with compute

        const float* w  = lds_w[ci & 1];
        const int    kc = ci * KC;
#pragma unroll
        for (int k = 0; k < KC; k += 4) {
            // fp32 WMMA A layout: VGPR0 = K0 (lanes 0-15) / K2 (16-31), VGPR1 = K1/K3.
            const int kl0 = k + 2 * half;
            const int kl1 = kl0 + 1;
            const int p0  = kc + kl0;
            const int p1  = kc + kl1;
            v2f zb;   // Z rows kl0/kl1, col d0+lm — shared by both WMMAs
            zb.x = xb0[(p0 >> LOG2H) * DF + d0 + lm] * xbx[(p0 & (H - 1)) * DF + d0 + lm];
            zb.y = xb0[(p1 >> LOG2H) * DF + d0 + lm] * xbx[(p1 & (H - 1)) * DF + d0 + lm];
            v2f a0, a1;
            a0.x = w[kl0 * O + o0 + lm];        a0.y = w[kl1 * O + o0 + lm];
            a1.x = w[kl0 * O + o0 + 16 + lm];   a1.y = w[kl1 * O + o0 + 16 + lm];
            c0 = __builtin_amdgcn_wmma_f32_16x16x4_f32(false, a0, false, zb,
                                                       (short)0, c0, false, false);
            c1 = __builtin_amdgcn_wmma_f32_16x16x4_f32(false, a1, false, zb,
                                                       (short)0, c1, false, false);
        }
        __syncthreads();         // (B) compute done before this buffer is reused
    }

    const int b = b0 + bb;

    // C/D layout: VGPR r -> row r (lanes 0-15) / row r+8 (lanes 16-31); col = lane%16.
    if (xout) {
#pragma unroll
        for (int r = 0; r < 8; ++r) {
            const int orow = o0 + r + 8 * half;
            const int col  = d0 + lm;
            xout[(size_t)b * O * DF + orow * DF + col]        = c0[r];
            xout[(size_t)b * O * DF + (orow + 16) * DF + col] = c1[r];
        }
    }

    // finals[b, foff+o] += sum_d row. 16-lane reduction (xor masks 1/2/4/8 stay
    // inside each 16-lane group on wave32); exactly two waves (d0 = 0,16) hit
    // each element -> two-way float atomicAdd is order-independent.
#pragma unroll
    for (int r = 0; r < 8; ++r) {
        float v = c0[r], u = c1[r];
        v += __shfl_xor(v, 8, 32);  u += __shfl_xor(u, 8, 32);
        v += __shfl_xor(v, 4, 32);  u += __shfl_xor(u, 4, 32);
        v += __shfl_xor(v, 2, 32);  u += __shfl_xor(u, 2, 32);
        v += __shfl_xor(v, 1, 32);  u += __shfl_xor(u, 1, 32);
        if (lm == 0) {
            const int orow = o0 + r + 8 * half;
            atomicAdd(&finals[(size_t)b * 224 + foff + orow],      v);
            atomicAdd(&finals[(size_t)b * 224 + foff + orow + 16], u);
        }
    }
}

// logit[b] = finals[b,:] . kf
__global__ __launch_bounds__(256)
void cin_final_kernel(const float* __restrict__ finals,
                      const float* __restrict__ kf,
                      float* __restrict__ out)
{
    __shared__ float red[256];
    const int b = blockIdx.x;
    const int t = threadIdx.x;
    float v = 0.f;
    if (t < 224) v = finals[(size_t)b * 224 + t] * kf[t];
    red[t] = v;
    __syncthreads();
#pragma unroll
    for (int s = 128; s > 0; s >>= 1) {
        if (t < s) red[t] += red[t + s];
        __syncthreads();
    }
    if (t == 0) out[b] = red[0];
}

extern "C" void kernel_launch(void* const* d_in, const int* in_sizes, int n_in,
                              void* d_out, int out_size, void* d_ws, size_t ws_size,
                              hipStream_t stream)
{
    const float* x0 = (const float*)d_in[0];   // [B,32,32]
    const float* k1 = (const float*)d_in[1];   // [1024,32]
    const float* k2 = (const float*)d_in[2];   // [1024,128]
    const float* k3 = (const float*)d_in[3];   // [4096,64]
    const float* kf = (const float*)d_in[4];   // [224]
    float* out = (float*)d_out;                // [B,1] fp32

    // Workspace: x1 (8 MB) | x2 (32 MB) | finals (1.75 MB)
    char*  ws  = (char*)d_ws;
    float* x1  = (float*)ws;
    float* x2  = (float*)(ws + (size_t)BATCH * 32  * 32 * sizeof(float));
    float* fin = (float*)(ws + (size_t)BATCH * 32  * 32 * sizeof(float)
                             + (size_t)BATCH * 128 * 32 * sizeof(float));

    (void)hipMemsetAsync(fin, 0, (size_t)BATCH * 224 * sizeof(float), stream);

    // Layer 1: H=32, O=32, xin==x0, 4 batches/block, KC=64. LDS 32 KB.
    cin_layer_kernel<32, 32, 5, 4, 64, true>
        <<<BATCH / 4, 256, 0, stream>>>(x0, x0, k1, x1, fin, 0);
    // Layer 2: H=32, O=128, 1 batch/block, KC=32. LDS 40 KB.
    cin_layer_kernel<32, 128, 5, 1, 32, false>
        <<<BATCH, 256, 0, stream>>>(x0, x1, k2, x2, fin, 32);
    // Layer 3: H=128, O=64, 2 batches/block, KC=32, x3 never stored. LDS 56 KB.
    cin_layer_kernel<128, 64, 7, 2, 32, false>
        <<<BATCH / 2, 256, 0, stream>>>(x0, x2, k3, nullptr, fin, 160);

    cin_final_kernel<<<BATCH, 256, 0, stream>>>(fin, kf, out);
}